// HeightmapRenderingLoss_48627619725707
// MI455X (gfx1250) — compile-verified
//
#include <hip/hip_runtime.h>

typedef float v2f __attribute__((ext_vector_type(2)));
typedef float v8f __attribute__((ext_vector_type(8)));

// w_a = 2^-32 * C(32, 16+a), a = -16..16  (binomial weights of a 32-step 1D walk)
#define WS 2.3283064365386963e-10f
__constant__ float WC[33] = {
  1.0f*WS, 32.0f*WS, 496.0f*WS, 4960.0f*WS, 35960.0f*WS, 201376.0f*WS,
  906192.0f*WS, 3365856.0f*WS, 10518300.0f*WS, 28048800.0f*WS,
  64512240.0f*WS, 129024480.0f*WS, 225792840.0f*WS, 347373600.0f*WS,
  471435600.0f*WS, 565722720.0f*WS, 601080390.0f*WS, 565722720.0f*WS,
  471435600.0f*WS, 347373600.0f*WS, 225792840.0f*WS, 129024480.0f*WS,
  64512240.0f*WS, 28048800.0f*WS, 10518300.0f*WS, 3365856.0f*WS,
  906192.0f*WS, 201376.0f*WS, 35960.0f*WS, 4960.0f*WS, 496.0f*WS,
  32.0f*WS, 1.0f*WS
};

// Tile geometry (output tile 64x64, halo 32 each side)
#define IMG     4096
#define TILE    64
#define INR     128          // input rows/cols per tile (TILE + 64)
#define XS_S    136          // Xs row stride (pad: 2*136 % 64 == 16 -> conflict-free dual-row loads)
#define TW      96           // t-region logical width/height (TILE + 32)
#define TS_S    104          // Ts stride (2*104 % 64 == 16)
#define TS2_S   108          // Ts2 stride (8*108 % 64 == 32)
#define XS_FL   (INR * XS_S)             // 17408 floats
#define TS_FL   (TW * TS_S)              // 9984 floats
#define STASH_FL (TILE * TILE)           // 4096 floats
#define LDS_FLOATS (XS_FL + TS_FL + STASH_FL)   // 31488 -> 125952 bytes

static __device__ __forceinline__ v8f wmma4(v2f a, v2f b, v8f c) {
  // D = A(16x4) * B(4x16) + C ; 8-arg form: (neg_a, A, neg_b, B, c_mod, C, reuse_a, reuse_b)
  return __builtin_amdgcn_wmma_f32_16x16x4_f32(false, a, false, b, (short)0, c,
                                               false, false);
}

// One 16x16 output block of a 33-tap vertical conv: Out = M0*B0 + M1*B1 + M2*B2
// (two accumulator chains to halve the serial WMMA dependency depth)
static __device__ __forceinline__ v8f conv_block(const v2f (&W)[3][4],
                                                 const v2f (&B0)[4],
                                                 const v2f (&B1)[4],
                                                 const v2f (&B2)[4]) {
  v8f a0 = {}, a1 = {};
  a0 = wmma4(W[0][0], B0[0], a0);  a1 = wmma4(W[2][0], B2[0], a1);
  a0 = wmma4(W[0][1], B0[1], a0);  a1 = wmma4(W[2][1], B2[1], a1);
  a0 = wmma4(W[0][2], B0[2], a0);  a1 = wmma4(W[2][2], B2[2], a1);
  a0 = wmma4(W[0][3], B0[3], a0);  a1 = wmma4(W[2][3], B2[3], a1);
  a0 = wmma4(W[1][0], B1[0], a0);  a1 = wmma4(W[1][2], B1[2], a1);
  a0 = wmma4(W[1][1], B1[1], a0);  a1 = wmma4(W[1][3], B1[3], a1);
  return a0 + a1;
}

__global__ __launch_bounds__(256)
void ao_render_kernel(const float* __restrict__ x,
                      const float* __restrict__ nrm,
                      float* __restrict__ out) {
  extern __shared__ float lds[];
  float* Xs    = lds;                     // 128 x 136 (shear-u input tile)
  float* Ts    = lds + XS_FL;             // 96 x 104  (shear-v stage-1 result)
  float* Stash = lds + XS_FL + TS_FL;     // 64 x 64   (central x values)
  float* Ts2   = lds;                     // 64 x 108  (stage-2 result, overlays Xs)

  const int tid  = threadIdx.x;
  const int wave = tid >> 5;
  const int lane = tid & 31;
  const int hl   = lane >> 4;     // lane-half (selects K pair / M+8)
  const int ln   = lane & 15;

  const int Y0 = blockIdx.y * TILE;
  const int X0 = blockIdx.x * TILE;

  // ---------------- load 128x128 reflected input tile into shear-u LDS ----------------
  for (int idx = tid; idx < INR * INR; idx += 256) {
    int row = idx >> 7;
    int cx  = idx & 127;
    int gy = Y0 - 32 + row;
    int gx = X0 - 32 + cx;
    gy = (gy < 0) ? (-1 - gy) : ((gy > IMG - 1) ? (2 * IMG - 1 - gy) : gy);
    gx = (gx < 0) ? (-1 - gx) : ((gx > IMG - 1) ? (2 * IMG - 1 - gx) : gx);
    float v = x[gy * IMG + gx];
    int cu = (row + cx) & 127;            // shear-u column (mod 128)
    Xs[row * XS_S + cu] = v;
    if (row >= 32 && row < 96 && cx >= 32 && cx < 96)
      Stash[(row - 32) * TILE + (cx - 32)] = v;
  }

  // ---------------- per-lane A-operand (band-matrix) fragments ----------------
  // A chunk (k,kc): value at (M=ln, K=2*hl+q), global band col j = 16k+4kc+2hl+q,
  // weight = WC[j - M] if 0<=j-M<=32 else 0.
  v2f W[3][4];
#pragma unroll
  for (int k = 0; k < 3; ++k)
#pragma unroll
    for (int kc = 0; kc < 4; ++kc) {
      int d0 = 16 * k + 4 * kc + 2 * hl - ln;
      float a0 = (d0 >= 0 && d0 <= 32) ? WC[d0] : 0.0f;
      float a1 = (d0 + 1 >= 0 && d0 + 1 <= 32) ? WC[d0 + 1] : 0.0f;
      v2f w; w.x = a0; w.y = a1;
      W[k][kc] = w;
    }

  __syncthreads();

  // ---------------- stage 1: 33-tap vertical conv in shear-u -> Ts (shear-v) ----------------
  {
    const int c = wave * 16 + ln;         // this wave's u-column strip
    v2f B0[4], B1[4], B2[4];
#pragma unroll
    for (int kc = 0; kc < 4; ++kc) {
      int rr = 4 * kc + 2 * hl;
      v2f t;
      t.x = Xs[(rr + 0) * XS_S + c];  t.y = Xs[(rr + 1) * XS_S + c];  B0[kc] = t;
      t.x = Xs[(rr + 16) * XS_S + c]; t.y = Xs[(rr + 17) * XS_S + c]; B1[kc] = t;
      t.x = Xs[(rr + 32) * XS_S + c]; t.y = Xs[(rr + 33) * XS_S + c]; B2[kc] = t;
    }
    for (int rb = 0; rb < 6; ++rb) {
      const int r0 = 16 + 16 * rb;        // output rows (input-row space)
      v8f acc = conv_block(W, B0, B1, B2);
      // scatter D into shear-v layout: t(ly, lx=c-ly) -> Ts[ty][ (tx-ty) mod 96 ]
#pragma unroll
      for (int g = 0; g < 8; ++g) {
        int ly = r0 + g + 8 * hl;
        int lx = c - ly; if (lx < 0) lx += 128;
        if (lx >= 16 && lx < 112) {
          int ty = ly - 16, tx = lx - 16;
          int dd = tx - ty; int cv = (dd < 0) ? dd + TW : dd;
          Ts[ty * TS_S + cv] = acc[g];
        }
      }
      if (rb < 5) {                       // slide 48-row window down by 16
#pragma unroll
        for (int kc = 0; kc < 4; ++kc) {
          B0[kc] = B1[kc];
          B1[kc] = B2[kc];
          int rr = r0 + 32 + 4 * kc + 2 * hl;
          v2f t;
          t.x = Xs[(rr + 0) * XS_S + c];
          t.y = Xs[(rr + 1) * XS_S + c];
          B2[kc] = t;
        }
      }
    }
  }

  __syncthreads();

  // ---------------- stage 2: 33-tap vertical conv in shear-v -> Ts2 (row-major (Y,v)) ----------------
  if (wave < 6) {
    const int c = wave * 16 + ln;         // v-column strip (96 columns over 6 waves)
    v2f B0[4], B1[4], B2[4];
#pragma unroll
    for (int kc = 0; kc < 4; ++kc) {
      int rr = 4 * kc + 2 * hl;
      v2f t;
      t.x = Ts[(rr + 0) * TS_S + c];  t.y = Ts[(rr + 1) * TS_S + c];  B0[kc] = t;
      t.x = Ts[(rr + 16) * TS_S + c]; t.y = Ts[(rr + 17) * TS_S + c]; B1[kc] = t;
      t.x = Ts[(rr + 32) * TS_S + c]; t.y = Ts[(rr + 33) * TS_S + c]; B2[kc] = t;
    }
    for (int rb = 0; rb < 4; ++rb) {
      const int r2 = 16 + 16 * rb;
      v8f acc = conv_block(W, B0, B1, B2);
#pragma unroll
      for (int g = 0; g < 8; ++g) {
        int row = r2 - 16 + g + 8 * hl;   // output tile row
        Ts2[row * TS2_S + c] = acc[g];
      }
      if (rb < 3) {
#pragma unroll
        for (int kc = 0; kc < 4; ++kc) {
          B0[kc] = B1[kc];
          B1[kc] = B2[kc];
          int rr = r2 + 32 + 4 * kc + 2 * hl;
          v2f t;
          t.x = Ts[(rr + 0) * TS_S + c];
          t.y = Ts[(rr + 1) * TS_S + c];
          B2[kc] = t;
        }
      }
    }
  }

  __syncthreads();

  // ---------------- epilogue: AO + diffuse shading, write output ----------------
  for (int idx = tid; idx < TILE * TILE; idx += 256) {
    int oy = idx >> 6;
    int ox = idx & 63;
    int dd = ox - oy; int cv = (dd < 0) ? dd + TW : dd;
    float rx = Ts2[oy * TS2_S + cv];
    float xv = Stash[idx];
    float ao = 1.0f - fminf(fmaxf((rx - xv) * 4.0f, 0.0f), 1.0f);
    int gidx = (Y0 + oy) * IMG + (X0 + ox);
    float n0 = nrm[gidx];
    float n1 = nrm[gidx + IMG * IMG];
    float n2 = nrm[gidx + 2 * IMG * IMG];
    float d1 = fmaxf(0.7053f * ((n0 - n1) + n2), 0.0f);
    out[gidx] = (0.3f * d1 + 0.7f) * ao;
  }
}

extern "C" void kernel_launch(void* const* d_in, const int* in_sizes, int n_in,
                              void* d_out, int out_size, void* d_ws, size_t ws_size,
                              hipStream_t stream) {
  (void)in_sizes; (void)n_in; (void)d_ws; (void)ws_size; (void)out_size;
  const float* x   = (const float*)d_in[0];
  const float* nrm = (const float*)d_in[1];
  float* out = (float*)d_out;
  dim3 grid(IMG / TILE, IMG / TILE);      // 64 x 64 tiles
  size_t ldsBytes = LDS_FLOATS * sizeof(float);  // 125,952 B
  ao_render_kernel<<<grid, dim3(256), ldsBytes, stream>>>(x, nrm, out);
}